// GCNFN_38457137168399
// MI455X (gfx1250) — compile-verified
//
#include <hip/hip_runtime.h>
#include <hip/hip_bf16.h>

#define N_NODES  100000
#define N_EDGES  1200000
#define E_TOT    1300000   // + self loops
#define IN_CH    128
#define HID      64
#define N_GRAPHS 256

typedef float v2f __attribute__((ext_vector_type(2)));
typedef float v8f __attribute__((ext_vector_type(8)));

// ---------- helpers ----------

// monotonic float->uint mapping so unsigned atomicMax == float max
__device__ __forceinline__ unsigned int fenc(float f) {
    unsigned int u = __float_as_uint(f);
    return (u & 0x80000000u) ? ~u : (u | 0x80000000u);
}
__device__ __forceinline__ float fdec(unsigned int u) {
    u = (u & 0x80000000u) ? (u ^ 0x80000000u) : ~u;
    return __uint_as_float(u);
}

__device__ __forceinline__ float selu_f(float v) {
    const float kA = 1.6732632423543772f;
    const float kS = 1.0507009873554805f;
    return v > 0.f ? kS * v : kS * kA * (__expf(v) - 1.f);
}

__device__ __forceinline__ void edge_sd(const int* __restrict__ ei, int e,
                                        int& s, int& d) {
    if (e < N_EDGES) { s = ei[e]; d = ei[N_EDGES + e]; }
    else             { s = e - N_EDGES; d = s; }   // self loop
}

// ---------- generic fill ----------
__global__ void fill_u32(unsigned int* __restrict__ p, unsigned int v, int n) {
    int i = blockIdx.x * 256 + threadIdx.x;
    if (i < n) p[i] = v;
}

// ---------- fp32 WMMA GEMM: C[M,64] = A[M,K] * B[K,64] ----------
// One wave computes a 16x64 tile via 4 accumulators of V_WMMA_F32_16X16X4_F32.
// B is staged in LDS PRE-SWIZZLED into the WMMA B-operand layout:
//   pair (k4, nt, lane) -> { B[(4*k4 + 2*(lane>>4))    *64 + nt*16 + (lane&15)],
//                            B[(4*k4 + 2*(lane>>4) + 1)*64 + nt*16 + (lane&15)] }
// so the inner loop issues one conflict-free ds_load_b64 per WMMA source
// (even-aligned VGPR pair, no repack movs).
__global__ void __launch_bounds__(256)
gat_wmma_gemm(const float* __restrict__ A, const float* __restrict__ B,
              float* __restrict__ C, int M, int K) {
    __shared__ float2 sB[(IN_CH / 4) * 128];          // 32KB max (K=128)
    const int tid = threadIdx.x;
    const int nPairs = (K / 4) * 128;
    for (int idx = tid; idx < nPairs; idx += 256) {
        int k4  = idx >> 7;
        int rem = idx & 127;
        int nt  = rem >> 5;
        int ln  = rem & 31;
        int row = k4 * 4 + (ln >> 4) * 2;
        int col = nt * 16 + (ln & 15);
        sB[idx] = make_float2(B[row * HID + col], B[(row + 1) * HID + col]);
    }
    __syncthreads();

    const int wave = tid >> 5;
    const int lane = tid & 31;
    const int half = lane >> 4;   // 0: K pair {0,1}, 1: K pair {2,3}
    const int lh   = lane & 15;
    const int m0   = blockIdx.x * 128 + wave * 16;
    if (m0 >= M) return;

    v8f acc[4];
#pragma unroll
    for (int nt = 0; nt < 4; ++nt)
#pragma unroll
        for (int r = 0; r < 8; ++r) acc[nt][r] = 0.f;

    // A layout (16x4 f32): lanes 0-15 -> M=lane, K={0,1}; lanes 16-31 -> M=lane-16, K={2,3}
    const float* arow = A + (size_t)(m0 + lh) * K;

    for (int k4 = 0; k4 < K / 4; ++k4) {
        float2 av = *(const float2*)(arow + k4 * 4 + half * 2);
        v2f a; a[0] = av.x; a[1] = av.y;
        const float2* bp = sB + k4 * 128 + lane;
#pragma unroll
        for (int nt = 0; nt < 4; ++nt) {
            float2 bv = bp[nt * 32];                  // single ds_load_b64
            v2f b; b[0] = bv.x; b[1] = bv.y;
            acc[nt] = __builtin_amdgcn_wmma_f32_16x16x4_f32(
                false, a, false, b, (short)0, acc[nt], false, false);
        }
    }

    // C/D layout: VGPR r -> row m0+r (lanes 0-15) / m0+8+r (lanes 16-31), col = nt*16+lh
#pragma unroll
    for (int nt = 0; nt < 4; ++nt)
#pragma unroll
        for (int r = 0; r < 8; ++r)
            C[(size_t)(m0 + half * 8 + r) * HID + nt * 16 + lh] = acc[nt][r];
}

// ---------- per-node attention scores: el = h.al, er = h.ar (wave per node) ----------
__global__ void __launch_bounds__(256)
node_scores(const float* __restrict__ h, const float* __restrict__ al,
            const float* __restrict__ ar, float* __restrict__ el,
            float* __restrict__ er, int n) {
    int w    = (blockIdx.x * 256 + threadIdx.x) >> 5;
    int lane = threadIdx.x & 31;
    if (w >= n) return;
    float2 hv  = *(const float2*)(h + (size_t)w * HID + lane * 2);
    float2 alv = *(const float2*)(al + lane * 2);
    float2 arv = *(const float2*)(ar + lane * 2);
    float pl = hv.x * alv.x + hv.y * alv.y;
    float pr = hv.x * arv.x + hv.y * arv.y;
#pragma unroll
    for (int off = 16; off > 0; off >>= 1) {
        pl += __shfl_xor(pl, off, 32);
        pr += __shfl_xor(pr, off, 32);
    }
    if (lane == 0) { el[w] = pl; er[w] = pr; }
}

// ---------- edge pass 1: e = leaky_relu(el[s]+er[d]); segment max ----------
__global__ void edge_score_max(const int* __restrict__ ei,
                               const float* __restrict__ el,
                               const float* __restrict__ er,
                               float* __restrict__ ea,
                               unsigned int* __restrict__ menc) {
    int e = blockIdx.x * 256 + threadIdx.x;
    if (e >= E_TOT) return;
    int s, d; edge_sd(ei, e, s, d);
    float t = el[s] + er[d];
    t = t > 0.f ? t : 0.2f * t;
    ea[e] = t;
    atomicMax(menc + d, fenc(t));
}

__global__ void decode_max(unsigned int* __restrict__ m, int n) {
    int i = blockIdx.x * 256 + threadIdx.x;
    if (i < n) m[i] = __float_as_uint(fdec(m[i]));
}

// ---------- edge pass 2: a = exp(e - m[d]); segment sum ----------
__global__ void edge_exp_sum(const int* __restrict__ ei,
                             const float* __restrict__ m,
                             float* __restrict__ ea, float* __restrict__ z) {
    int e = blockIdx.x * 256 + threadIdx.x;
    if (e >= E_TOT) return;
    int s, d; edge_sd(ei, e, s, d);
    float a = __expf(ea[e] - m[d]);
    ea[e] = a;
    atomicAdd(z + d, a);
}

// ---------- edge pass 3: agg[d] += (a/z[d]) * h[s]  (wave per edge, 2 ch/lane) ----------
__global__ void __launch_bounds__(256)
edge_aggregate(const int* __restrict__ ei, const float* __restrict__ ea,
               const float* __restrict__ z, const float* __restrict__ h,
               float* __restrict__ agg) {
    int g = blockIdx.x * 256 + threadIdx.x;
    int e = g >> 5, lane = g & 31;
    if (e >= E_TOT) return;
    int s, d; edge_sd(ei, e, s, d);
    float alpha = ea[e] / z[d];
    float2 hv = *(const float2*)(h + (size_t)s * HID + lane * 2);
    atomicAdd(agg + (size_t)d * HID + lane * 2,     alpha * hv.x);
    atomicAdd(agg + (size_t)d * HID + lane * 2 + 1, alpha * hv.y);
}

// ---------- epilogue: h = selu(agg + b) in place ----------
__global__ void selu_bias(float* __restrict__ hb, const float* __restrict__ b,
                          int total) {
    int i = blockIdx.x * 256 + threadIdx.x;
    if (i >= total) return;
    hb[i] = selu_f(hb[i] + b[i & (HID - 1)]);
}

// ---------- mean pooling accumulation ----------
__global__ void pool_accum(const float* __restrict__ h, const int* __restrict__ batch,
                           float* __restrict__ pooled, float* __restrict__ cnt) {
    int i = blockIdx.x * 256 + threadIdx.x;
    if (i >= N_NODES * HID) return;
    int n = i >> 6, c = i & 63;
    int g = batch[n];
    atomicAdd(pooled + (size_t)g * HID + c, h[i]);
    if (c == 0) atomicAdd(cnt + g, 1.f);
}

// ---------- MLP head + log_softmax (one thread per graph) ----------
__global__ void __launch_bounds__(256)
head_mlp(const float* __restrict__ pooled, const float* __restrict__ cnt,
         const float* __restrict__ w1, const float* __restrict__ b1,
         const float* __restrict__ w2, const float* __restrict__ b2,
         float* __restrict__ out) {
    int g = threadIdx.x;   // 1 block, 256 threads
    float c = cnt[g]; c = c > 1.f ? c : 1.f;
    float inv = 1.f / c;
    float gv[HID];
#pragma unroll 8
    for (int i = 0; i < HID; ++i) gv[i] = pooled[(size_t)g * HID + i] * inv;
    float h1[HID / 2];
    for (int j = 0; j < HID / 2; ++j) {
        float a = b1[j];
        for (int i = 0; i < HID; ++i) a += gv[i] * w1[i * (HID / 2) + j];
        h1[j] = selu_f(a);
    }
    float o0 = b2[0], o1 = b2[1];
    for (int j = 0; j < HID / 2; ++j) { o0 += h1[j] * w2[j * 2]; o1 += h1[j] * w2[j * 2 + 1]; }
    float mx  = fmaxf(o0, o1);
    float lse = mx + __logf(__expf(o0 - mx) + __expf(o1 - mx));
    out[g * 2]     = o0 - lse;
    out[g * 2 + 1] = o1 - lse;
}

// ---------- orchestration ----------
extern "C" void kernel_launch(void* const* d_in, const int* in_sizes, int n_in,
                              void* d_out, int out_size, void* d_ws, size_t ws_size,
                              hipStream_t stream) {
    const float* x     = (const float*)d_in[0];
    const int*   ei    = (const int*)  d_in[1];
    const int*   batch = (const int*)  d_in[2];
    const float* W1    = (const float*)d_in[3];
    const float* al1   = (const float*)d_in[4];
    const float* ar1   = (const float*)d_in[5];
    const float* b1    = (const float*)d_in[6];
    const float* W2    = (const float*)d_in[7];
    const float* al2   = (const float*)d_in[8];
    const float* ar2   = (const float*)d_in[9];
    const float* b2    = (const float*)d_in[10];
    const float* fc1w  = (const float*)d_in[11];
    const float* fc1b  = (const float*)d_in[12];
    const float* fc2w  = (const float*)d_in[13];
    const float* fc2b  = (const float*)d_in[14];

    // workspace layout (floats): two ping-pong feature buffers + small arrays
    float* bufA   = (float*)d_ws;                       // h (GEMM output)        [N*64]
    float* bufB   = bufA + (size_t)N_NODES * HID;       // agg / selu / next in   [N*64]
    float* el     = bufB + (size_t)N_NODES * HID;       // [N]
    float* er     = el + N_NODES;                       // [N]
    float* mbuf   = er + N_NODES;                       // [N] (uint-encoded then decoded)
    float* zbuf   = mbuf + N_NODES;                     // [N]
    float* ea     = zbuf + N_NODES;                     // [E_TOT]
    float* pooled = ea + E_TOT;                         // [G*64]
    float* cnt    = pooled + (size_t)N_GRAPHS * HID;    // [G]

    const dim3 blk(256);
    const int gemmGrid = (N_NODES + 127) / 128;
    const int nodeWav  = (N_NODES * 32 + 255) / 256;
    const int nGrid    = (N_NODES + 255) / 256;
    const int edgeGrid = (E_TOT + 255) / 256;
    const int eaggGrid = (E_TOT * 32 + 255) / 256;
    const int nhGrid   = (N_NODES * HID + 255) / 256;
    const unsigned int ENC_NEG_INF = 0x007FFFFFu;       // fenc(-inf)

    // ---- layer 1 ----
    gat_wmma_gemm<<<gemmGrid, blk, 0, stream>>>(x, W1, bufA, N_NODES, IN_CH);
    node_scores<<<nodeWav, blk, 0, stream>>>(bufA, al1, ar1, el, er, N_NODES);
    fill_u32<<<nGrid,  blk, 0, stream>>>((unsigned int*)mbuf, ENC_NEG_INF, N_NODES);
    fill_u32<<<nGrid,  blk, 0, stream>>>((unsigned int*)zbuf, 0u, N_NODES);
    fill_u32<<<nhGrid, blk, 0, stream>>>((unsigned int*)bufB, 0u, N_NODES * HID);
    edge_score_max<<<edgeGrid, blk, 0, stream>>>(ei, el, er, ea, (unsigned int*)mbuf);
    decode_max<<<nGrid, blk, 0, stream>>>((unsigned int*)mbuf, N_NODES);
    edge_exp_sum<<<edgeGrid, blk, 0, stream>>>(ei, mbuf, ea, zbuf);
    edge_aggregate<<<eaggGrid, blk, 0, stream>>>(ei, ea, zbuf, bufA, bufB);
    selu_bias<<<nhGrid, blk, 0, stream>>>(bufB, b1, N_NODES * HID);

    // ---- layer 2 ----
    gat_wmma_gemm<<<gemmGrid, blk, 0, stream>>>(bufB, W2, bufA, N_NODES, HID);
    node_scores<<<nodeWav, blk, 0, stream>>>(bufA, al2, ar2, el, er, N_NODES);
    fill_u32<<<nGrid,  blk, 0, stream>>>((unsigned int*)mbuf, ENC_NEG_INF, N_NODES);
    fill_u32<<<nGrid,  blk, 0, stream>>>((unsigned int*)zbuf, 0u, N_NODES);
    fill_u32<<<nhGrid, blk, 0, stream>>>((unsigned int*)bufB, 0u, N_NODES * HID);
    edge_score_max<<<edgeGrid, blk, 0, stream>>>(ei, el, er, ea, (unsigned int*)mbuf);
    decode_max<<<nGrid, blk, 0, stream>>>((unsigned int*)mbuf, N_NODES);
    edge_exp_sum<<<edgeGrid, blk, 0, stream>>>(ei, mbuf, ea, zbuf);
    edge_aggregate<<<eaggGrid, blk, 0, stream>>>(ei, ea, zbuf, bufA, bufB);
    selu_bias<<<nhGrid, blk, 0, stream>>>(bufB, b2, N_NODES * HID);

    // ---- mean pool + MLP head ----
    fill_u32<<<(N_GRAPHS * HID + N_GRAPHS + 255) / 256, blk, 0, stream>>>(
        (unsigned int*)pooled, 0u, N_GRAPHS * HID + N_GRAPHS);
    pool_accum<<<nhGrid, blk, 0, stream>>>(bufB, batch, pooled, cnt);
    head_mlp<<<1, blk, 0, stream>>>(pooled, cnt, fc1w, fc1b, fc2w, fc2b, (float*)d_out);
}